// DKEncoder_35029753266372
// MI455X (gfx1250) — compile-verified
//
#include <hip/hip_runtime.h>
#include <hip/hip_bf16.h>
#include <math.h>

typedef float v2f __attribute__((ext_vector_type(2)));
typedef float v4f __attribute__((ext_vector_type(4)));
typedef float v8f __attribute__((ext_vector_type(8)));

#define Bsz 8
#define Ssz 256
#define Esz 256
#define Qd  768
#define Dd  100
#define N1  16
#define N2  16

// ---------------------------------------------------------------------------
// Kernel 1: w = (tanh(q0 @ Wq^T + bq)) @ Wk   for one layer, one block.
// Two chained fp32 WMMA GEMMs (V_WMMA_F32_16X16X4_F32).
// 7 waves, each owning one 16-wide N tile (7*16 = 112 >= 100).
// Out-of-range lanes read a clamped (valid) row instead of being predicated:
// garbage A-rows / B-columns only pollute discarded C rows/columns, so the
// K-loop body is pure {2x global_load_b64 + v_wmma} with no EXEC masking.
// ---------------------------------------------------------------------------
__global__ __launch_bounds__(224) void qproj_wmma(const float* __restrict__ q,
                                                  const float* __restrict__ Wq,
                                                  const float* __restrict__ bq,
                                                  const float* __restrict__ Wk,
                                                  float* __restrict__ w_out) {
  __shared__ float qi_s[16 * Dd];
  const int t    = threadIdx.x;
  const int wave = t >> 5;
  const int lane = t & 31;
  const int n    = lane & 15;
  const int h    = lane >> 4;     // half: selects K pair {2h, 2h+1}
  const int m    = lane & 15;     // A row held by this lane
  const int nn   = wave * 16 + n; // global N (column) index

  for (int i = t; i < 16 * Dd; i += 224) qi_s[i] = 0.0f;

  // Clamped row pointers: unconditional loads, junk lands in discarded tiles.
  const float* qrow = q  + (size_t)((m  < Bsz) ? m  : 0) * Ssz * Qd; // q[:,0,:]
  const float* wrow = Wq + (size_t)((nn < Dd ) ? nn : 0) * Qd;
  const int    ncol = (nn < Dd) ? nn : 0;

  // GEMM1: qi[m, nn] = sum_k q0[m,k] * Wq[nn,k]   (M=16 pad of 8, K=768)
  v8f c = {};
  for (int k0 = 0; k0 < Qd; k0 += 4) {
    v2f a;
    a.x = qrow[k0 + 2 * h];
    a.y = qrow[k0 + 2 * h + 1];
    v2f b;
    b.x = wrow[k0 + 2 * h];
    b.y = wrow[k0 + 2 * h + 1];
    c = __builtin_amdgcn_wmma_f32_16x16x4_f32(false, a, false, b, (short)0, c,
                                              false, false);
  }
  __syncthreads();
  if (h == 0 && nn < Dd) {
    const float bias = bq[nn];
#pragma unroll
    for (int i = 0; i < 8; ++i) qi_s[i * Dd + nn] = tanhf(c[i] + bias);
  }
  __syncthreads();

  // GEMM2: w[m, nn] = sum_k qi[m,k] * Wk[k,nn]   (K=100)
  v8f c2 = {};
  for (int k0 = 0; k0 < Dd; k0 += 4) {
    v2f a;
    a.x = qi_s[m * Dd + k0 + 2 * h];
    a.y = qi_s[m * Dd + k0 + 2 * h + 1];
    v2f b;
    b.x = Wk[(size_t)(k0 + 2 * h) * Dd + ncol];
    b.y = Wk[(size_t)(k0 + 2 * h + 1) * Dd + ncol];
    c2 = __builtin_amdgcn_wmma_f32_16x16x4_f32(false, a, false, b, (short)0, c2,
                                               false, false);
  }
  if (h == 0 && nn < Dd) {
#pragma unroll
    for (int i = 0; i < 8; ++i) w_out[i * Dd + nn] = c2[i];
  }
}

// ---------------------------------------------------------------------------
// Kernel 2: fused two-layer CokeBert attention for one (b,e) per block.
// Streams k2/v2/k1/v1 exactly once with NT cache hints ->
// bandwidth-roofline (~446 MB @ 23.3 TB/s ~= 19 us floor).
// ---------------------------------------------------------------------------
__global__ __launch_bounds__(256) void dk_main(const float* __restrict__ k1,
                                               const float* __restrict__ v1,
                                               const float* __restrict__ k2,
                                               const float* __restrict__ v2,
                                               const float* __restrict__ wvec,
                                               float* __restrict__ comb1) {
  const int g = blockIdx.x;           // b*E + e
  const int b = g >> 8;               // E == 256
  const int t = threadIdx.x;

  __shared__ float w2s[Dd], w1s[Dd];
  __shared__ float lg[256], att[256];
  __shared__ float c2s[N1 * Dd];
  __shared__ float lg1[N1], att1[N1];

  if (t < Dd) {
    w2s[t] = wvec[b * Dd + t];
    w1s[t] = wvec[Bsz * Dd + b * Dd + t];
  }
  __syncthreads();

  // ---- layer 2: logits over 256 (n1,n2) rows, one row per thread ----
  const size_t gb2 = (size_t)g * (N1 * N2 * Dd);
  {
    const v4f* krow = (const v4f*)(k2 + gb2 + (size_t)t * Dd);
    float acc = 0.0f;
#pragma unroll
    for (int i = 0; i < Dd / 4; ++i) {
      const v4f kk = __builtin_nontemporal_load(krow + i);
      acc += kk.x * w2s[4 * i] + kk.y * w2s[4 * i + 1] +
             kk.z * w2s[4 * i + 2] + kk.w * w2s[4 * i + 3];
    }
    float logit = acc * 0.1f;                 // 1/sqrt(100)
    if (logit == 0.0f) logit = -10000.0f;
    lg[t] = logit > 0.0f ? logit : 0.01f * logit;   // leaky_relu
  }
  __syncthreads();
  {
    const int rb = t & ~15;                   // this thread's n1-row base
    float mx = -INFINITY;
#pragma unroll
    for (int j = 0; j < N2; ++j) mx = fmaxf(mx, lg[rb + j]);
    float se = 0.0f;
#pragma unroll
    for (int j = 0; j < N2; ++j) se += __expf(lg[rb + j] - mx);
    float a = __expf(lg[t] - mx) / se;
    if (a == 0.0625f) a = 0.0f;               // uniform => all-masked row
    att[t] = a;
  }
  __syncthreads();

  // ---- combined2[n1][d] = sum_n2 att * v2 ----
  for (int idx = t; idx < N1 * Dd; idx += 256) {
    const int n1 = idx / Dd;
    const int d  = idx - n1 * Dd;
    const float* vb = v2 + gb2 + (size_t)n1 * (N2 * Dd) + d;
    float s = 0.0f;
#pragma unroll
    for (int n2 = 0; n2 < N2; ++n2)
      s += att[n1 * 16 + n2] * __builtin_nontemporal_load(vb + n2 * Dd);
    c2s[idx] = s;
  }
  __syncthreads();

  // ---- layer 1: logits over n1 ----
  const size_t gb1 = (size_t)g * (N1 * Dd);
  if (t < N1) {
    const v4f* krow = (const v4f*)(k1 + gb1 + (size_t)t * Dd);
    float acc = 0.0f;
#pragma unroll
    for (int i = 0; i < Dd / 4; ++i) {
      const v4f kk = __builtin_nontemporal_load(krow + i);
      acc += kk.x * w1s[4 * i] + kk.y * w1s[4 * i + 1] +
             kk.z * w1s[4 * i + 2] + kk.w * w1s[4 * i + 3];
    }
    float logit = acc * 0.1f;
    if (logit == 0.0f) logit = -10000.0f;
    lg1[t] = logit > 0.0f ? logit : 0.01f * logit;
  }
  __syncthreads();
  if (t < N1) {
    float mx = -INFINITY;
#pragma unroll
    for (int j = 0; j < N1; ++j) mx = fmaxf(mx, lg1[j]);
    float se = 0.0f;
#pragma unroll
    for (int j = 0; j < N1; ++j) se += __expf(lg1[j] - mx);
    float a = __expf(lg1[t] - mx) / se;
    if (a == 0.0625f) a = 0.0f;
    att1[t] = a;
  }
  __syncthreads();

  // ---- combined1[d] over concat([v1, combined2]) ----
  if (t < 2 * Dd) {
    float s = 0.0f;
    if (t < Dd) {
      const float* vb = v1 + gb1 + t;
#pragma unroll
      for (int n1 = 0; n1 < N1; ++n1)
        s += att1[n1] * __builtin_nontemporal_load(vb + n1 * Dd);
    } else {
      const int d = t - Dd;
#pragma unroll
      for (int n1 = 0; n1 < N1; ++n1) s += att1[n1] * c2s[n1 * Dd + d];
    }
    comb1[(size_t)g * (2 * Dd) + t] = s;
  }
}

// ---------------------------------------------------------------------------
// Kernel 3: rank = clip(cumsum(mask)-1), masked gather of 200-wide rows.
// One block per batch; Hillis-Steele inclusive scan over S=256.
// ---------------------------------------------------------------------------
__global__ __launch_bounds__(256) void gather_out(const int* __restrict__ input_ent,
                                                  const float* __restrict__ comb1,
                                                  float* __restrict__ out) {
  const int b = blockIdx.x;
  const int t = threadIdx.x;
  __shared__ int sc[256];
  __shared__ int rk[256];
  __shared__ int mk[256];

  const int m = (input_ent[b * Ssz + t] != 0) ? 1 : 0;
  mk[t] = m;
  sc[t] = m;
  __syncthreads();
  for (int off = 1; off < 256; off <<= 1) {
    int v = sc[t];
    if (t >= off) v += sc[t - off];
    __syncthreads();
    sc[t] = v;
    __syncthreads();
  }
  int rank = sc[t] - 1;
  rank = rank < 0 ? 0 : (rank > Esz - 1 ? Esz - 1 : rank);
  rk[t] = rank;
  __syncthreads();

  for (int idx = t; idx < Ssz * 2 * Dd; idx += 256) {
    const int s = idx / (2 * Dd);
    const int d = idx - s * (2 * Dd);
    out[((size_t)b * Ssz + s) * (2 * Dd) + d] =
        mk[s] ? comb1[((size_t)b * Esz + rk[s]) * (2 * Dd) + d] : 0.0f;
  }
}

// ---------------------------------------------------------------------------
extern "C" void kernel_launch(void* const* d_in, const int* in_sizes, int n_in,
                              void* d_out, int out_size, void* d_ws, size_t ws_size,
                              hipStream_t stream) {
  const int*   input_ent = (const int*)d_in[0];   // jax default: int64 -> int32
  const float* q   = (const float*)d_in[1];
  const float* k1  = (const float*)d_in[2];
  const float* v1  = (const float*)d_in[3];
  const float* k2  = (const float*)d_in[4];
  const float* v2  = (const float*)d_in[5];
  const float* Wq2 = (const float*)d_in[6];
  const float* bq2 = (const float*)d_in[7];
  const float* Wk2 = (const float*)d_in[8];
  const float* Wq1 = (const float*)d_in[9];
  const float* bq1 = (const float*)d_in[10];
  const float* Wk1 = (const float*)d_in[11];
  float* out = (float*)d_out;
  float* wsf = (float*)d_ws;

  float* w2    = wsf;                       // [8,100]
  float* w1    = wsf + Bsz * Dd;            // [8,100]
  float* comb1 = wsf + 2 * Bsz * Dd;        // [8,256,200]

  qproj_wmma<<<1, 224, 0, stream>>>(q, Wq2, bq2, Wk2, w2);
  qproj_wmma<<<1, 224, 0, stream>>>(q, Wq1, bq1, Wk1, w1);
  dk_main<<<Bsz * Esz, 256, 0, stream>>>(k1, v1, k2, v2, wsf, comb1);
  gather_out<<<Bsz, 256, 0, stream>>>(input_ent, comb1, out);
}